// LongTermSpectralFlatness_22771916603503
// MI455X (gfx1250) — compile-verified
//
#include <hip/hip_runtime.h>

// ---------------------------------------------------------------------------
// LongTermSpectralFlatness on MI455X (gfx1250, wave32).
//
// welch / am / glog are banded linear operators over time -> implemented as
// chained V_WMMA_F32_16X16X4_F32 (fp32 matrix pipe), with the band
// coefficients (1/min(t,M) inside [max(0,t-M), t), else 0) generated per-lane.
// Memory-bound problem (~154 MB in / 23.3 TB/s ~= 6.6 us floor); input staged
// once per tile into LDS with a 42-frame halo, halo re-reads hit the 192 MB L2.
// ---------------------------------------------------------------------------

typedef __attribute__((ext_vector_type(2))) float v2f;
typedef __attribute__((ext_vector_type(8))) float v8f;

#define T_FRAMES 3000
#define F_BINS   201
#define B_BATCH  32
#define TT       64                 // output frames per block
#define HALO     42                 // s halo: 30 (R window) + 10 (M window) + 2 align
#define SROWS    (TT + HALO)        // 106 staged frames
#define FP       208                // padded freq bins (13 chunks of 16)
#define NCH      13
#define EPSF     1e-5f
// hamming_sq_sum(25)/SAMPLE_RATE/M = 9.935/16000/10
#define SCONST   6.209375e-05f
#define INV_LN10 0.4342944819032518f

// welch window coefficient: mean over s[max(0,t-10) .. t), row t==0 -> 0
__device__ __forceinline__ float coefA(int t, int u) {
  bool in = (t >= 1) & (u >= 0) & (u >= t - 10) & (u < t);
  int cnt = (t < 10) ? (t > 0 ? t : 1) : 10;
  return in ? (1.0f / (float)cnt) : 0.0f;
}
// long-term window coefficient: mean over welch[max(0,t-30) .. t)
__device__ __forceinline__ float coefV(int t, int u) {
  bool in = (t >= 1) & (u >= 0) & (u >= t - 30) & (u < t);
  int cnt = (t < 30) ? (t > 0 ? t : 1) : 30;
  return in ? (1.0f / (float)cnt) : 0.0f;
}

__global__ __launch_bounds__(256, 1)
void ltsf_kernel(const float* __restrict__ x, float* __restrict__ out) {
  __shared__ float sL[SROWS * FP];        // staged scaled power spectrum (88 KB)
  __shared__ float wsc[8 * 48 * 16];      // per-wave welch scratch (24 KB)
  __shared__ float lsc[8 * 48 * 16];      // per-wave log-welch scratch (24 KB)
  __shared__ float red[8][16];            // cross-wave flatness partials

  const int b   = blockIdx.y;
  const int t0  = blockIdx.x * TT;
  const int tid = threadIdx.x;

  // ---- Phase 1: stage s[u][f] = (re^2+im^2) * SCONST * scale[f] into LDS ----
  for (int idx = tid; idx < SROWS * FP; idx += 256) {
    int r = idx / FP, f = idx - r * FP;
    int u = t0 - HALO + r;
    float v = 0.0f;
    if (f < F_BINS && u >= 0 && u < T_FRAMES) {
      const float2 ri =
          *(const float2*)(x + (((size_t)b * T_FRAMES + u) * F_BINS + f) * 2);
      float sc = (f == 0 || f == F_BINS - 1) ? SCONST : 2.0f * SCONST;
      v = (ri.x * ri.x + ri.y * ri.y) * sc;
    }
    sL[idx] = v;
  }
  __syncthreads();

  const int lane = tid & 31;
  const int wave = tid >> 5;
  const int sub  = wave & 3;           // which 16-frame subtile this wave owns
  const int tsub = t0 + 16 * sub;
  const int tb   = tsub - 32;          // base of 48-row welch block
  const int m    = lane & 15;          // A-matrix row / C-matrix column
  const int koff = (lane >> 4) << 1;   // K offset per lane-half: 0 | 2
  const int crow = (lane >> 4) << 3;   // C-matrix row offset per half: 0 | 8
  float* wp = wsc + wave * (48 * 16);
  float* lp = lsc + wave * (48 * 16);

  v8f acc = {0.f, 0.f, 0.f, 0.f, 0.f, 0.f, 0.f, 0.f};

  // waves 0-3: even freq chunks of subtile (wave&3); waves 4-7: odd chunks
  for (int ch = (wave >> 2); ch < NCH; ch += 2) {
    const int f0 = ch * 16;

    // ---- welch = A(16x[band]) x s via 7 chained K=4 WMMAs per 16-row tile ----
    #pragma unroll
    for (int i = 0; i < 3; ++i) {
      const int ta = tb + 16 * i;
      v8f c = {0.f, 0.f, 0.f, 0.f, 0.f, 0.f, 0.f, 0.f};
      #pragma unroll
      for (int j = 0; j < 7; ++j) {
        const int ub = ta - 10 + 4 * j;   // band covers u in [ta-10, ta+15]
        v2f a, bv;
        a.x = coefA(ta + m, ub + koff);
        a.y = coefA(ta + m, ub + koff + 1);
        int su0 = ub + koff - (t0 - HALO);
        int su1 = su0 + 1;
        su0 = su0 < 0 ? 0 : (su0 > SROWS - 1 ? SROWS - 1 : su0);
        su1 = su1 < 0 ? 0 : (su1 > SROWS - 1 ? SROWS - 1 : su1);
        bv.x = sL[su0 * FP + f0 + m];
        bv.y = sL[su1 * FP + f0 + m];
        c = __builtin_amdgcn_wmma_f32_16x16x4_f32(
            false, a, false, bv, (short)0, c, false, false);
      }
      // stash welch and log(welch+eps) in B-layout-addressable LDS scratch
      #pragma unroll
      for (int e = 0; e < 8; ++e) {
        const int row = 16 * i + e + crow;   // C layout: VGPR e -> row e | e+8
        const float w = c[e];
        wp[row * 16 + m] = w;
        lp[row * 16 + m] = __logf(w + EPSF);
      }
    }

    // ---- am = V x welch, glog = V x log(welch+eps): 12 K=4 WMMAs each ----
    v8f amc = {0.f, 0.f, 0.f, 0.f, 0.f, 0.f, 0.f, 0.f};
    v8f glc = {0.f, 0.f, 0.f, 0.f, 0.f, 0.f, 0.f, 0.f};
    #pragma unroll
    for (int k = 0; k < 12; ++k) {
      const int ubl = 4 * k;          // local row in the 48-row welch block
      const int u   = tb + ubl;       // absolute frame of row ubl
      v2f a, bw, bl;
      a.x = coefV(tsub + m, u + koff);
      a.y = coefV(tsub + m, u + koff + 1);
      const int r0 = ubl + koff;
      bw.x = wp[r0 * 16 + m];
      bw.y = wp[(r0 + 1) * 16 + m];
      bl.x = lp[r0 * 16 + m];
      bl.y = lp[(r0 + 1) * 16 + m];
      amc = __builtin_amdgcn_wmma_f32_16x16x4_f32(
          false, a, false, bw, (short)0, amc, false, false);
      glc = __builtin_amdgcn_wmma_f32_16x16x4_f32(
          false, a, false, bl, (short)0, glc, false, false);
    }

    // ---- epilogue: -log10(gm/am), accumulate over this freq chunk ----
    #pragma unroll
    for (int e = 0; e < 8; ++e) {
      const int t = tsub + e + crow;
      const float am = amc[e] + EPSF;
      const float gm = (__expf(glc[e]) - EPSF) + EPSF;  // mirrors source +/-eps
      float val = -(__logf(gm) - __logf(am)) * INV_LN10;
      if (t <= 0 || t >= T_FRAMES || (f0 + m) >= F_BINS) val = 0.0f;
      acc[e] += val;
    }
  }

  // ---- reduce the 16 frequency lanes per half-wave (butterfly) ----
  #pragma unroll
  for (int e = 0; e < 8; ++e) {
    float v = acc[e];
    v += __shfl_xor(v, 1, 32);
    v += __shfl_xor(v, 2, 32);
    v += __shfl_xor(v, 4, 32);
    v += __shfl_xor(v, 8, 32);
    acc[e] = v;
  }
  if (lane == 0) {
    #pragma unroll
    for (int e = 0; e < 8; ++e) red[wave][e] = acc[e];
  }
  if (lane == 16) {
    #pragma unroll
    for (int e = 0; e < 8; ++e) red[wave][e + 8] = acc[e];
  }
  __syncthreads();

  // ---- combine the two waves per subtile, store flatness (B,T,1) ----
  if (tid < 64) {
    const int s2 = tid >> 4, i2 = tid & 15;
    const int t = t0 + 16 * s2 + i2;
    if (t < T_FRAMES)
      out[(size_t)b * T_FRAMES + t] = red[s2][i2] + red[s2 + 4][i2];
  }
}

extern "C" void kernel_launch(void* const* d_in, const int* in_sizes, int n_in,
                              void* d_out, int out_size, void* d_ws,
                              size_t ws_size, hipStream_t stream) {
  (void)in_sizes; (void)n_in; (void)d_ws; (void)ws_size; (void)out_size;
  const float* x = (const float*)d_in[0];
  float* out = (float*)d_out;
  dim3 grid((T_FRAMES + TT - 1) / TT, B_BATCH);
  ltsf_kernel<<<grid, dim3(256), 0, stream>>>(x, out);
}